// MovementGATModel_pos_83141976916258
// MI455X (gfx1250) — compile-verified
//
#include <hip/hip_runtime.h>

typedef float v2f __attribute__((ext_vector_type(2)));
typedef float v8f __attribute__((ext_vector_type(8)));

#define HC 256   // H*C
#define NH 4     // heads
#define CPH 64   // channels per head

// ---- ordered-uint encoding so atomicMax(unsigned) == float max ----
__device__ __forceinline__ unsigned enc_ord(float f) {
  unsigned u = __float_as_uint(f);
  return (u & 0x80000000u) ? ~u : (u | 0x80000000u);
}
__device__ __forceinline__ float dec_ord(unsigned u) {
  return __uint_as_float((u & 0x80000000u) ? (u ^ 0x80000000u) : ~u);
}
__device__ __forceinline__ float lrelu02(float x) { return x > 0.f ? x : 0.2f * x; }

// ---------------- small elementwise / setup kernels ----------------
__global__ void k_zero(float* p, int n) {
  int i = blockIdx.x * 256 + threadIdx.x;
  if (i < n) p[i] = 0.f;
}

__global__ void k_edge_feat(const int* __restrict__ src, const int* __restrict__ dst,
                            const float* __restrict__ pos, float* __restrict__ agg, int E) {
  int e = blockIdx.x * 256 + threadIdx.x;
  if (e >= E) return;
  int s = src[e], t = dst[e];
  float dx = pos[t * 2 + 0] - pos[s * 2 + 0];
  float dy = pos[t * 2 + 1] - pos[s * 2 + 1];
  atomicAdd(&agg[s * 2 + 0], sqrtf(dx * dx + dy * dy));
  atomicAdd(&agg[s * 2 + 1], atan2f(dy, dx));
}

__global__ void k_build_xf(const float* __restrict__ x, const float* __restrict__ agg,
                           float* __restrict__ xf, int N, int Fin) {
  int Kin = Fin + 2;
  int i = blockIdx.x * 256 + threadIdx.x;
  if (i >= N * Kin) return;
  int n = i / Kin, c = i % Kin;
  xf[i] = (c < Fin) ? x[n * Fin + c] : agg[n * 2 + (c - Fin)];
}

// ---------------- WMMA f32 GEMM: C[M,Ncol] = A[M,K] @ B[K,Ncol] ----------------
// One wave per 16x16 output tile; V_WMMA_F32_16X16X4_F32, K stepped by 4.
__global__ void k_gemm_wmma(const float* __restrict__ A, const float* __restrict__ B,
                            float* __restrict__ C, int M, int Ncol, int K) {
  int gid  = blockIdx.x * blockDim.x + threadIdx.x;
  int wave = gid >> 5;
  int lane = threadIdx.x & 31;
  int tnc  = Ncol >> 4;
  int tm = wave / tnc, tn = wave % tnc;
  if (tm * 16 >= M) return;               // uniform per-wave: EXEC stays all-1s
  int m    = tm * 16 + (lane & 15);       // A row for this lane
  int n    = tn * 16 + (lane & 15);       // B/C column for this lane
  int koff = (lane >> 4) << 1;            // lane group -> K pair {0,1} or {2,3}
  v8f acc = {0.f, 0.f, 0.f, 0.f, 0.f, 0.f, 0.f, 0.f};
  for (int k = 0; k < K; k += 4) {
    v2f a, b;
    a.x = A[m * K + (k + koff)];
    a.y = A[m * K + (k + koff + 1)];
    b.x = B[(k + koff) * Ncol + n];
    b.y = B[(k + koff + 1) * Ncol + n];
    acc = __builtin_amdgcn_wmma_f32_16x16x4_f32(false, a, false, b, (short)0, acc,
                                                false, false);
  }
  int mb = tm * 16 + (lane >> 4) * 8;     // C: VGPR r -> row (lane>=16?8:0)+r
#pragma unroll
  for (int r = 0; r < 8; ++r) C[(mb + r) * Ncol + n] = acc[r];
}

// ---------------- attention ----------------
__global__ void k_attn_scal(const float* __restrict__ h, const float* __restrict__ a_src,
                            const float* __restrict__ a_dst, float* __restrict__ s_sc,
                            float* __restrict__ d_sc, int N) {
  int i = blockIdx.x * 256 + threadIdx.x;  // over N*NH
  if (i >= N * NH) return;
  int n = i >> 2, hh = i & 3;
  const float* hp = h + n * HC + hh * CPH;
  const float* as = a_src + hh * CPH;
  const float* ad = a_dst + hh * CPH;
  float ss = 0.f, dd = 0.f;
  for (int c = 0; c < CPH; ++c) { float v = hp[c]; ss += v * as[c]; dd += v * ad[c]; }
  s_sc[i] = ss; d_sc[i] = dd;
}

__global__ void k_init_attn(unsigned* __restrict__ m_enc, float* __restrict__ sumE,
                            float* __restrict__ oacc, int N) {
  int i = blockIdx.x * 256 + threadIdx.x;
  if (i < N * HC) oacc[i] = 0.f;
  if (i < N * NH) { m_enc[i] = 0x007FFFFFu /* enc(-inf) */; sumE[i] = 0.f; }
}

__global__ void k_edge_max(const int* __restrict__ src, const int* __restrict__ dst,
                           const float* __restrict__ s_sc, const float* __restrict__ d_sc,
                           unsigned* __restrict__ m_enc, int E, int N) {
  int idx = blockIdx.x * 256 + threadIdx.x;  // over (E+N)*NH
  if (idx >= (E + N) * NH) return;
  int e = idx >> 2, hh = idx & 3;
  int s, t;
  if (e < E) { s = src[e]; t = dst[e]; } else { s = e - E; t = s; }  // self-loops
  float logit = lrelu02(s_sc[s * NH + hh] + d_sc[t * NH + hh]);
  atomicMax(&m_enc[t * NH + hh], enc_ord(logit));
}

__global__ void k_edge_acc(const int* __restrict__ src, const int* __restrict__ dst,
                           const float* __restrict__ s_sc, const float* __restrict__ d_sc,
                           const unsigned* __restrict__ m_enc, const float* __restrict__ h,
                           float* __restrict__ sumE, float* __restrict__ oacc, int E, int N) {
  long long idx = (long long)blockIdx.x * 256 + threadIdx.x;  // over (E+N)*HC
  if (idx >= (long long)(E + N) * HC) return;
  int e = (int)(idx >> 8);
  int j = (int)(idx & 255);
  int hh = j >> 6;
  int s, t;
  if (e < E) { s = src[e]; t = dst[e]; } else { s = e - E; t = s; }
  float logit = lrelu02(s_sc[s * NH + hh] + d_sc[t * NH + hh]);
  float ee = __expf(logit - dec_ord(m_enc[t * NH + hh]));
  atomicAdd(&oacc[t * HC + j], ee * h[s * HC + j]);
  if ((j & 63) == 0) atomicAdd(&sumE[t * NH + hh], ee);
}

__global__ void k_norm_bias(const float* __restrict__ oacc, const float* __restrict__ sumE,
                            const float* __restrict__ b, float* __restrict__ xnorm, int N) {
  int i = blockIdx.x * 256 + threadIdx.x;
  if (i >= N * HC) return;
  int n = i >> 8, j = i & 255, hh = j >> 6;
  xnorm[i] = oacc[i] / (sumE[n * NH + hh] + 1e-16f) + b[j];
}

// ---------------- batchnorm (train-mode, biased var) + relu ----------------
__global__ void k_bn_stats(const float* __restrict__ x, float* __restrict__ mu,
                           float* __restrict__ rstd, int N) {
  __shared__ float s1[256], s2[256];
  int j = blockIdx.x, t = threadIdx.x;
  float a = 0.f, b = 0.f;
  for (int i = t; i < N; i += 256) { float v = x[i * HC + j]; a += v; b += v * v; }
  s1[t] = a; s2[t] = b;
  __syncthreads();
  for (int st = 128; st > 0; st >>= 1) {
    if (t < st) { s1[t] += s1[t + st]; s2[t] += s2[t + st]; }
    __syncthreads();
  }
  if (t == 0) {
    float m = s1[0] / (float)N;
    float var = s2[0] / (float)N - m * m;
    mu[j] = m;
    rstd[j] = rsqrtf(var + 1e-5f);
  }
}

__global__ void k_bn_apply(const float* __restrict__ x, const float* __restrict__ mu,
                           const float* __restrict__ rstd, const float* __restrict__ g,
                           const float* __restrict__ be, float* __restrict__ y, int N) {
  int i = blockIdx.x * 256 + threadIdx.x;
  if (i >= N * HC) return;
  int j = i & 255;
  float v = g[j] * (x[i] - mu[j]) * rstd[j] + be[j];
  y[i] = v > 0.f ? v : 0.f;
}

// ---------------- final projection + mask + pos update ----------------
__global__ void k_final(const float* __restrict__ h, const float* __restrict__ Wf,
                        const float* __restrict__ bf, const float* __restrict__ mask,
                        const float* __restrict__ pos, float* __restrict__ out, int N) {
  int n = blockIdx.x * 256 + threadIdx.x;
  if (n >= N) return;
  float y0 = bf[0], y1 = bf[1];
  const float* hp = h + n * HC;
  for (int j = 0; j < HC; ++j) { float v = hp[j]; y0 += v * Wf[j * 2]; y1 += v * Wf[j * 2 + 1]; }
  float mk = mask[n];
  y0 *= mk; y1 *= mk;
  out[n * 2 + 0] = y0;
  out[n * 2 + 1] = y1;
  out[N * 2 + n * 2 + 0] = pos[n * 2 + 0] + y0;
  out[N * 2 + n * 2 + 1] = pos[n * 2 + 1] + y1;
}

// ---------------- one GAT layer ----------------
static void run_gat_layer(const float* Xin, int Kin, const float* W, const float* as,
                          const float* ad, const float* b, const float* g, const float* be,
                          const int* src, const int* dst, int E, int N, float* hbuf,
                          float* s_sc, float* d_sc, unsigned* m_enc, float* sumE,
                          float* oacc, float* xnorm, float* mu, float* rstd, float* hout,
                          hipStream_t stream) {
  int tiles = (N / 16) * (HC / 16);
  k_gemm_wmma<<<(tiles * 32 + 255) / 256, 256, 0, stream>>>(Xin, W, hbuf, N, HC, Kin);
  k_attn_scal<<<(N * NH + 255) / 256, 256, 0, stream>>>(hbuf, as, ad, s_sc, d_sc, N);
  k_init_attn<<<(N * HC + 255) / 256, 256, 0, stream>>>(m_enc, sumE, oacc, N);
  int Etot = E + N;
  k_edge_max<<<(Etot * NH + 255) / 256, 256, 0, stream>>>(src, dst, s_sc, d_sc, m_enc, E, N);
  long long accth = (long long)Etot * HC;
  k_edge_acc<<<(unsigned)((accth + 255) / 256), 256, 0, stream>>>(src, dst, s_sc, d_sc,
                                                                  m_enc, hbuf, sumE, oacc, E, N);
  k_norm_bias<<<(N * HC + 255) / 256, 256, 0, stream>>>(oacc, sumE, b, xnorm, N);
  k_bn_stats<<<HC, 256, 0, stream>>>(xnorm, mu, rstd, N);
  k_bn_apply<<<(N * HC + 255) / 256, 256, 0, stream>>>(xnorm, mu, rstd, g, be, hout, N);
}

extern "C" void kernel_launch(void* const* d_in, const int* in_sizes, int n_in,
                              void* d_out, int out_size, void* d_ws, size_t ws_size,
                              hipStream_t stream) {
  const float* x    = (const float*)d_in[0];
  const int*   ei   = (const int*)d_in[1];
  const float* pos  = (const float*)d_in[2];
  const float* mask = (const float*)d_in[3];
  /* d_in[4] = batch, unused by reference */
  const float* W1  = (const float*)d_in[5];
  const float* as1 = (const float*)d_in[6];
  const float* ad1 = (const float*)d_in[7];
  const float* b1  = (const float*)d_in[8];
  const float* g1  = (const float*)d_in[9];
  const float* be1 = (const float*)d_in[10];
  const float* W2  = (const float*)d_in[11];
  const float* as2 = (const float*)d_in[12];
  const float* ad2 = (const float*)d_in[13];
  const float* b2  = (const float*)d_in[14];
  const float* g2  = (const float*)d_in[15];
  const float* be2 = (const float*)d_in[16];
  const float* Wf  = (const float*)d_in[17];
  const float* bf  = (const float*)d_in[18];

  int N   = in_sizes[2] / 2;     // 2560
  int E   = in_sizes[1] / 2;     // 327680
  int Fin = in_sizes[0] / N;     // 62
  int Kin = Fin + 2;             // 64
  const int* src = ei;
  const int* dst = ei + E;

  float* w = (float*)d_ws;
  float* agg   = w; w += (size_t)N * 2;
  float* xf    = w; w += (size_t)N * Kin;
  float* hbuf  = w; w += (size_t)N * HC;
  float* s_sc  = w; w += (size_t)N * NH;
  float* d_sc  = w; w += (size_t)N * NH;
  unsigned* m_enc = (unsigned*)w; w += (size_t)N * NH;
  float* sumE  = w; w += (size_t)N * NH;
  float* oacc  = w; w += (size_t)N * HC;
  float* xnorm = w; w += (size_t)N * HC;
  float* hact1 = w; w += (size_t)N * HC;
  float* hact2 = w; w += (size_t)N * HC;
  float* mu    = w; w += 256;
  float* rstd  = w; w += 256;

  k_zero<<<(N * 2 + 255) / 256, 256, 0, stream>>>(agg, N * 2);
  k_edge_feat<<<(E + 255) / 256, 256, 0, stream>>>(src, dst, pos, agg, E);
  k_build_xf<<<(N * Kin + 255) / 256, 256, 0, stream>>>(x, agg, xf, N, Fin);

  run_gat_layer(xf, Kin, W1, as1, ad1, b1, g1, be1, src, dst, E, N, hbuf, s_sc, d_sc,
                m_enc, sumE, oacc, xnorm, mu, rstd, hact1, stream);
  run_gat_layer(hact1, HC, W2, as2, ad2, b2, g2, be2, src, dst, E, N, hbuf, s_sc, d_sc,
                m_enc, sumE, oacc, xnorm, mu, rstd, hact2, stream);

  k_final<<<(N + 255) / 256, 256, 0, stream>>>(hact2, Wf, bf, mask, pos, (float*)d_out, N);
}